// GumbelCustering_1460288881071
// MI455X (gfx1250) — compile-verified
//
#include <hip/hip_runtime.h>
#include <math.h>

#define Bx 1024
#define Dx 512
#define Kx 1024
#define Hx 32
#define PI_D 3.14159265358979323846
#define MIN_TAU 1e-08
#define MAX_TAU 10.0

typedef __attribute__((ext_vector_type(2))) float v2f;
typedef __attribute__((ext_vector_type(8))) float v8f;
typedef __attribute__((ext_vector_type(4))) int v4i;

#define AS1 __attribute__((address_space(1)))
#define AS3 __attribute__((address_space(3)))

// ---- CDNA5 async global->LDS (B128), with safe fallback ----
__device__ __forceinline__ void async_copy_b128(const float* g, float* lds) {
#if __has_builtin(__builtin_amdgcn_global_load_async_to_lds_b128)
    __builtin_amdgcn_global_load_async_to_lds_b128(
        (AS1 v4i*)(uintptr_t)g,
        (AS3 v4i*)(unsigned)(uintptr_t)lds,
        0 /*imm offset*/, 0 /*cpol*/);
#else
    *reinterpret_cast<float4*>(lds) = *reinterpret_cast<const float4*>(g);
#endif
}

__device__ __forceinline__ void wait_async_lds() {
#if __has_builtin(__builtin_amdgcn_global_load_async_to_lds_b128)
#if __has_builtin(__builtin_amdgcn_s_wait_asynccnt)
    __builtin_amdgcn_s_wait_asynccnt(0);
#else
    asm volatile("s_wait_asynccnt 0x0" ::: "memory");
#endif
#endif
}

// ---------------- init: argmin keys + delta accumulator ----------------
__global__ void init_kernel(unsigned long long* __restrict__ rowkey,
                            float* __restrict__ acc) {
    int i = blockIdx.x * 256 + threadIdx.x;
    if (i < Bx) rowkey[i] = ~0ull;
    if (i == 0) *acc = 0.f;
}

// ---------------- row norms of x and codebook ----------------
__global__ __launch_bounds__(256) void norms_kernel(const float* __restrict__ x,
                                                    const float* __restrict__ cb,
                                                    float* __restrict__ xn,
                                                    float* __restrict__ cn) {
    __shared__ float red[256];
    int row = blockIdx.x;
    const float* src;
    float* dst;
    int r;
    if (row < Bx) { src = x;  dst = xn; r = row; }
    else          { src = cb; dst = cn; r = row - Bx; }
    float s = 0.f;
    for (int d = threadIdx.x; d < Dx; d += 256) {
        float v = src[(size_t)r * Dx + d];
        s = fmaf(v, v, s);
    }
    red[threadIdx.x] = s;
    __syncthreads();
    for (int st = 128; st > 0; st >>= 1) {
        if (threadIdx.x < st) red[threadIdx.x] += red[threadIdx.x + st];
        __syncthreads();
    }
    if (threadIdx.x == 0) dst[r] = red[0];
}

// ---------------- h = relu(x @ w1 + b1)  (WMMA f32 16x16x4) ----------------
__global__ __launch_bounds__(32) void mlp1_kernel(const float* __restrict__ x,
                                                  const float* __restrict__ w1,
                                                  const float* __restrict__ b1,
                                                  float* __restrict__ h) {
    const int m0 = blockIdx.x * 16, n0 = blockIdx.y * 16;
    const int lane = threadIdx.x;
    const int half = lane >> 4, lm = lane & 15;
    v8f c = {};
    const float* xrow = x + (size_t)(m0 + lm) * Dx;
    for (int k = 0; k < Dx; k += 4) {
        const int kh = k + half * 2;
        v2f a, b;
        a.x = xrow[kh];
        a.y = xrow[kh + 1];
        b.x = w1[(size_t)kh * Hx + (n0 + lm)];
        b.y = w1[(size_t)(kh + 1) * Hx + (n0 + lm)];
        c = __builtin_amdgcn_wmma_f32_16x16x4_f32(false, a, false, b,
                                                  (short)0, c, false, false);
    }
    const float bias = b1[n0 + lm];
#pragma unroll
    for (int r = 0; r < 8; ++r) {
        const int m = m0 + r + 8 * half;
        h[(size_t)m * Hx + (n0 + lm)] = fmaxf(c[r] + bias, 0.f);
    }
}

// ---------------- logits = h @ w2 + b2  (WMMA f32 16x16x4) ----------------
__global__ __launch_bounds__(32) void mlp2_kernel(const float* __restrict__ h,
                                                  const float* __restrict__ w2,
                                                  const float* __restrict__ b2,
                                                  float* __restrict__ logits) {
    const int m0 = blockIdx.x * 16, n0 = blockIdx.y * 16;
    const int lane = threadIdx.x;
    const int half = lane >> 4, lm = lane & 15;
    v8f c = {};
    const float* hrow = h + (size_t)(m0 + lm) * Hx;
#pragma unroll
    for (int k = 0; k < Hx; k += 4) {
        const int kh = k + half * 2;
        v2f a, b;
        a.x = hrow[kh];
        a.y = hrow[kh + 1];
        b.x = w2[(size_t)kh * Kx + (n0 + lm)];
        b.y = w2[(size_t)(kh + 1) * Kx + (n0 + lm)];
        c = __builtin_amdgcn_wmma_f32_16x16x4_f32(false, a, false, b,
                                                  (short)0, c, false, false);
    }
    const float bias = b2[n0 + lm];
#pragma unroll
    for (int r = 0; r < 8; ++r) {
        const int m = m0 + r + 8 * half;
        logits[(size_t)m * Kx + (n0 + lm)] = c[r] + bias;
    }
}

// ---------------- z = softmax((logits + gumbel)/temperature) ----------------
__global__ __launch_bounds__(256) void softmax_kernel(const float* __restrict__ logits,
                                                      const float* __restrict__ gum,
                                                      const int* __restrict__ tptr,
                                                      float* __restrict__ z) {
    __shared__ float red[256];
    const int b = blockIdx.x;
    const int tid = threadIdx.x;
    const double tt = (double)(*tptr);
    double temp;
    if (MAX_TAU > tt)
        temp = MIN_TAU + 0.5 * (MAX_TAU - MIN_TAU) * (1.0 + cos(tt / MAX_TAU * PI_D));
    else
        temp = MIN_TAU;
    const float invT = (float)(1.0 / temp);

    const float* lrow = logits + (size_t)b * Kx;
    const float* grow = gum + (size_t)b * Kx;
    float m = -INFINITY;
    for (int k = tid; k < Kx; k += 256)
        m = fmaxf(m, (lrow[k] + grow[k]) * invT);
    red[tid] = m;
    __syncthreads();
    for (int s = 128; s > 0; s >>= 1) {
        if (tid < s) red[tid] = fmaxf(red[tid], red[tid + s]);
        __syncthreads();
    }
    m = red[0];
    __syncthreads();
    float sum = 0.f;
    for (int k = tid; k < Kx; k += 256)
        sum += expf((lrow[k] + grow[k]) * invT - m);
    red[tid] = sum;
    __syncthreads();
    for (int s = 128; s > 0; s >>= 1) {
        if (tid < s) red[tid] += red[tid + s];
        __syncthreads();
    }
    const float inv = 1.f / red[0];
    for (int k = tid; k < Kx; k += 256)
        z[(size_t)b * Kx + k] = expf((lrow[k] + grow[k]) * invT - m) * inv;
}

// ------- squared distance via WMMA (||x||^2 - 2 x.c + ||c||^2) + argmin -------
__global__ __launch_bounds__(32) void gemm_argmin_kernel(const float* __restrict__ x,
                                                         const float* __restrict__ cb,
                                                         const float* __restrict__ xn,
                                                         const float* __restrict__ cn,
                                                         unsigned long long* __restrict__ rowkey) {
    const int m0 = blockIdx.x * 16, n0 = blockIdx.y * 16;
    const int lane = threadIdx.x;
    const int half = lane >> 4, lm = lane & 15;
    v8f c = {};
    const float* xrow = x + (size_t)(m0 + lm) * Dx;
    const float* crow = cb + (size_t)(n0 + lm) * Dx;  // B = codebook^T: column n is cb row n
    for (int k = 0; k < Dx; k += 4) {
        const int kh = k + half * 2;
        v2f a, b;
        a.x = xrow[kh];
        a.y = xrow[kh + 1];
        b.x = crow[kh];
        b.y = crow[kh + 1];
        c = __builtin_amdgcn_wmma_f32_16x16x4_f32(false, a, false, b,
                                                  (short)0, c, false, false);
    }
    const float cnv = cn[n0 + lm];
#pragma unroll
    for (int r = 0; r < 8; ++r) {
        const int m = m0 + r + 8 * half;
        const float dist = xn[m] - 2.f * c[r] + cnv;  // >= 0 => fp bits are order-preserving
        unsigned long long key =
            ((unsigned long long)__float_as_uint(dist) << 32) | (unsigned)(n0 + lm);
        // min across the 16 lanes of this half (rows differ between halves)
#pragma unroll
        for (int msk = 8; msk >= 1; msk >>= 1) {
            unsigned long long o = __shfl_xor(key, msk, 32);
            if (o < key) key = o;
        }
        if (lm == 0) atomicMin(&rowkey[m], key);
    }
}

// ---------------- L1 distance * z -> delta (VALU-bound core) ----------------
// x tile DMA'd via GLOBAL_LOAD_ASYNC_TO_LDS_B128 (reads are lane-broadcast, so
// the unpadded 16-float rows cause no bank conflicts). Codebook tile keeps the
// padded synchronous path (its reads hit 16 distinct rows per warp and need +1
// padding, which is incompatible with 16B-contiguous async writes).
#define BT 64
#define KT 64
#define DC 16
__global__ __launch_bounds__(256) void l1_delta_kernel(const float* __restrict__ x,
                                                       const float* __restrict__ cb,
                                                       const float* __restrict__ z,
                                                       float* __restrict__ acc) {
    __shared__ float xs[BT][DC];       // async-filled, broadcast reads
    __shared__ float cs[KT][DC + 1];   // padded, conflict-free strided reads
    __shared__ float redf[256];
    const int t = threadIdx.x;
    const int b0 = blockIdx.x * BT;
    const int k0 = blockIdx.y * KT;
    const int tb = t >> 4;  // 0..15  -> 4 b-rows each
    const int tk = t & 15;  // 0..15  -> 4 k-cols each
    const int lr = t >> 2;  // 0..63  loader row
    const int lq = t & 3;   // 0..3   loader quad

    float s1[4][4];
#pragma unroll
    for (int i = 0; i < 4; ++i)
#pragma unroll
        for (int j = 0; j < 4; ++j) s1[i][j] = 0.f;

    for (int d0 = 0; d0 < Dx; d0 += DC) {
        // x tile: async DMA straight into LDS (no VGPR staging)
        async_copy_b128(&x[(size_t)(b0 + lr) * Dx + d0 + lq * 4], &xs[lr][lq * 4]);
        // codebook tile: synchronous, padded
        const float4 cv4 =
            *reinterpret_cast<const float4*>(&cb[(size_t)(k0 + lr) * Dx + d0 + lq * 4]);
        cs[lr][lq * 4 + 0] = cv4.x; cs[lr][lq * 4 + 1] = cv4.y;
        cs[lr][lq * 4 + 2] = cv4.z; cs[lr][lq * 4 + 3] = cv4.w;
        wait_async_lds();   // ASYNCcnt==0 before signaling the barrier
        __syncthreads();
#pragma unroll
        for (int d = 0; d < DC; ++d) {
            float xv[4], cv[4];
#pragma unroll
            for (int i = 0; i < 4; ++i) xv[i] = xs[tb * 4 + i][d];
#pragma unroll
            for (int j = 0; j < 4; ++j) cv[j] = cs[tk * 4 + j][d];
#pragma unroll
            for (int i = 0; i < 4; ++i)
#pragma unroll
                for (int j = 0; j < 4; ++j) s1[i][j] += fabsf(cv[j] - xv[i]);
        }
        __syncthreads();
    }

    float part = 0.f;
#pragma unroll
    for (int i = 0; i < 4; ++i)
#pragma unroll
        for (int j = 0; j < 4; ++j)
            part += s1[i][j] * z[(size_t)(b0 + tb * 4 + i) * Kx + (k0 + tk * 4 + j)];
    redf[t] = part;
    __syncthreads();
    for (int s = 128; s > 0; s >>= 1) {
        if (t < s) redf[t] += redf[t + s];
        __syncthreads();
    }
    if (t == 0) atomicAdd(acc, redf[0]);
}

// ---------------- finalize: indices + scaled delta ----------------
__global__ void finalize_kernel(const unsigned long long* __restrict__ rowkey,
                                const float* __restrict__ acc,
                                float* __restrict__ out) {
    int b = blockIdx.x * 256 + threadIdx.x;
    if (b < Bx)
        out[(size_t)Bx * Kx + b] = (float)(unsigned)(rowkey[b] & 0xFFFFFFFFull);
    if (b == 0)
        out[(size_t)Bx * Kx + Bx] =
            acc[0] * (1.0f / ((float)Bx * (float)Kx * (float)Dx));  // exact pow2
}

extern "C" void kernel_launch(void* const* d_in, const int* in_sizes, int n_in,
                              void* d_out, int out_size, void* d_ws, size_t ws_size,
                              hipStream_t stream) {
    const float* x   = (const float*)d_in[0];
    const float* cb  = (const float*)d_in[1];
    const float* w1  = (const float*)d_in[2];
    const float* b1  = (const float*)d_in[3];
    const float* w2  = (const float*)d_in[4];
    const float* b2  = (const float*)d_in[5];
    const float* gum = (const float*)d_in[6];
    const int*   tp  = (const int*)d_in[7];
    float* out = (float*)d_out;

    // workspace layout (floats): xn[Bx] | cn[Kx] | h[Bx*Hx] | z[Bx*Kx] | rowkey[Bx](u64) | acc
    float* xn = (float*)d_ws;
    float* cn = xn + Bx;
    float* h  = cn + Kx;
    float* zw = h + (size_t)Bx * Hx;
    unsigned long long* rowkey = (unsigned long long*)(zw + (size_t)Bx * Kx);
    float* acc = (float*)(rowkey + Bx);

    init_kernel<<<(Bx + 255) / 256, 256, 0, stream>>>(rowkey, acc);
    norms_kernel<<<Bx + Kx, 256, 0, stream>>>(x, cb, xn, cn);
    mlp1_kernel<<<dim3(Bx / 16, Hx / 16), 32, 0, stream>>>(x, w1, b1, h);
    mlp2_kernel<<<dim3(Bx / 16, Kx / 16), 32, 0, stream>>>(h, w2, b2, out);
    softmax_kernel<<<Bx, 256, 0, stream>>>(out, gum, tp, zw);
    gemm_argmin_kernel<<<dim3(Bx / 16, Kx / 16), 32, 0, stream>>>(x, cb, xn, cn, rowkey);
    l1_delta_kernel<<<dim3(Bx / BT, Kx / KT), 256, 0, stream>>>(x, cb, zw, acc);
    finalize_kernel<<<(Bx + 255) / 256, 256, 0, stream>>>(rowkey, acc, out);
}